// MR_GNN_76690936037562
// MI455X (gfx1250) — compile-verified
//
#include <hip/hip_runtime.h>
#include <hip/hip_bf16.h>

typedef __attribute__((ext_vector_type(2))) float v2f;
typedef __attribute__((ext_vector_type(8))) float v8f;

#define HD 64
#define NREL 4

static __device__ __forceinline__ v8f wmma4(v2f a, v2f b, v8f c) {
  // D = A(16x4 f32) * B(4x16 f32) + C(16x16 f32)
  return __builtin_amdgcn_wmma_f32_16x16x4_f32(false, a, false, b, (short)0, c,
                                               false, false);
}

// ---------------- degree ----------------
__global__ void k_degree(const int* __restrict__ ei, float* __restrict__ deg, int E) {
  int e = blockIdx.x * blockDim.x + threadIdx.x;
  if (e < E) atomicAdd(&deg[ei[E + e]], 1.0f);
}

__global__ void k_invdeg(float* __restrict__ deg, int N) {
  int i = blockIdx.x * blockDim.x + threadIdx.x;
  if (i < N) deg[i] = 1.0f / fmaxf(deg[i], 1.0f);
}

// ------- per-node relation transforms: Y[r] = X @ Wrel[r] (r<4), Y[4] = X @ W0 -------
__global__ void k_xform5(const float* __restrict__ X, const float* __restrict__ WREL,
                         const float* __restrict__ W0, float* __restrict__ Y, int N) {
  int gwave = (blockIdx.x * blockDim.x + threadIdx.x) >> 5;
  int lane  = threadIdx.x & 31;
  int tile  = gwave / 5;
  int mat   = gwave - tile * 5;
  int m0    = tile << 4;
  if (m0 >= N) return;                       // wave-uniform guard (EXEC all-1s for WMMA)
  const float* B = (mat < 4) ? (WREL + (size_t)mat * HD * HD) : W0;
  float* OUT = Y + (size_t)mat * N * HD;
  int mrow = lane & 15, half = lane >> 4;
  int ar = m0 + mrow; if (ar >= N) ar = N - 1;
  const float* arow = X + (size_t)ar * HD;
  v8f acc0 = {}, acc1 = {}, acc2 = {}, acc3 = {};
#pragma unroll
  for (int k = 0; k < HD; k += 4) {
    int kb = k + half * 2;
    v2f a  = { arow[kb], arow[kb + 1] };
    v2f b0 = { B[kb * HD + mrow],      B[(kb + 1) * HD + mrow] };
    v2f b1 = { B[kb * HD + 16 + mrow], B[(kb + 1) * HD + 16 + mrow] };
    v2f b2 = { B[kb * HD + 32 + mrow], B[(kb + 1) * HD + 32 + mrow] };
    v2f b3 = { B[kb * HD + 48 + mrow], B[(kb + 1) * HD + 48 + mrow] };
    acc0 = wmma4(a, b0, acc0);
    acc1 = wmma4(a, b1, acc1);
    acc2 = wmma4(a, b2, acc2);
    acc3 = wmma4(a, b3, acc3);
  }
#pragma unroll
  for (int v = 0; v < 8; v++) {
    int m = m0 + half * 8 + v;
    if (m < N) {
      float* orow = OUT + (size_t)m * HD;
      orow[mrow]      = acc0[v];
      orow[16 + mrow] = acc1[v];
      orow[32 + mrow] = acc2[v];
      orow[48 + mrow] = acc3[v];
    }
  }
}

// ------- relational scatter: agg[dst] += w * Y[edge_type][src] -------
__global__ void k_scatter_rel(const int* __restrict__ ei, const int* __restrict__ et,
                              const float* __restrict__ ew, const float* __restrict__ Y,
                              float* __restrict__ agg, int E, int N) {
  int t = blockIdx.x * blockDim.x + threadIdx.x;
  int e = t >> 4;
  if (e >= E) return;
  int c = (t & 15) << 2;
  int s = ei[e], d = ei[E + e], r = et[e];
  float w = ew[e];
  const float4 v = *(const float4*)(Y + ((size_t)r * N + s) * HD + c);
  float* dd = agg + (size_t)d * HD + c;
  atomicAdd(dd + 0, w * v.x);
  atomicAdd(dd + 1, w * v.y);
  atomicAdd(dd + 2, w * v.z);
  atomicAdd(dd + 3, w * v.w);
}

// ------- plain scatter: agg[dst] += w * Hin[src] -------
__global__ void k_scatter_plain(const int* __restrict__ ei, const float* __restrict__ ew,
                                const float* __restrict__ Hin, float* __restrict__ agg,
                                int E) {
  int t = blockIdx.x * blockDim.x + threadIdx.x;
  int e = t >> 4;
  if (e >= E) return;
  int c = (t & 15) << 2;
  int s = ei[e], d = ei[E + e];
  float w = ew[e];
  const float4 v = *(const float4*)(Hin + (size_t)s * HD + c);
  float* dd = agg + (size_t)d * HD + c;
  atomicAdd(dd + 0, w * v.x);
  atomicAdd(dd + 1, w * v.y);
  atomicAdd(dd + 2, w * v.z);
  atomicAdd(dd + 3, w * v.w);
}

// ------- RGCN epilogue: h = relu((S + agg) * invdeg) -------
__global__ void k_rgcn_finalize(const float* __restrict__ S, const float* __restrict__ AGG,
                                const float* __restrict__ INV, float* __restrict__ Hout,
                                int N) {
  int i = blockIdx.x * blockDim.x + threadIdx.x;
  if (i >= N * HD) return;
  float v = (S[i] + AGG[i]) * INV[i >> 6];
  Hout[i] = v > 0.f ? v : 0.f;
}

// ------- GroupEnhance: out = h + alpha * ((agg * invdeg) @ PW + PB) -------
__global__ void k_group(const float* __restrict__ Hin, const float* __restrict__ AGG,
                        const float* __restrict__ INV, const float* __restrict__ PW,
                        const float* __restrict__ PB, const float* __restrict__ ALPHA,
                        float* __restrict__ OUT, int N) {
  int gwave = (blockIdx.x * blockDim.x + threadIdx.x) >> 5;
  int lane  = threadIdx.x & 31;
  int m0    = gwave << 4;
  if (m0 >= N) return;
  int mrow = lane & 15, half = lane >> 4;
  int ar = m0 + mrow; if (ar >= N) ar = N - 1;
  const float* arow = AGG + (size_t)ar * HD;
  float inv = INV[ar];
  v8f acc0 = {}, acc1 = {}, acc2 = {}, acc3 = {};
#pragma unroll
  for (int k = 0; k < HD; k += 4) {
    int kb = k + half * 2;
    v2f a  = { arow[kb] * inv, arow[kb + 1] * inv };
    v2f b0 = { PW[kb * HD + mrow],      PW[(kb + 1) * HD + mrow] };
    v2f b1 = { PW[kb * HD + 16 + mrow], PW[(kb + 1) * HD + 16 + mrow] };
    v2f b2 = { PW[kb * HD + 32 + mrow], PW[(kb + 1) * HD + 32 + mrow] };
    v2f b3 = { PW[kb * HD + 48 + mrow], PW[(kb + 1) * HD + 48 + mrow] };
    acc0 = wmma4(a, b0, acc0);
    acc1 = wmma4(a, b1, acc1);
    acc2 = wmma4(a, b2, acc2);
    acc3 = wmma4(a, b3, acc3);
  }
  float alpha = ALPHA[0];
#pragma unroll
  for (int v = 0; v < 8; v++) {
    int m = m0 + half * 8 + v;
    if (m < N) {
      const float* hrow = Hin + (size_t)m * HD;
      float* orow = OUT + (size_t)m * HD;
      orow[mrow]      = hrow[mrow]      + alpha * (acc0[v] + PB[mrow]);
      orow[16 + mrow] = hrow[16 + mrow] + alpha * (acc1[v] + PB[16 + mrow]);
      orow[32 + mrow] = hrow[32 + mrow] + alpha * (acc2[v] + PB[32 + mrow]);
      orow[48 + mrow] = hrow[48 + mrow] + alpha * (acc3[v] + PB[48 + mrow]);
    }
  }
}

// ------- output head: out = h @ outW(64x32) + outb -------
__global__ void k_outhead(const float* __restrict__ Hin, const float* __restrict__ OW,
                          const float* __restrict__ OB, float* __restrict__ OUT, int N) {
  int gwave = (blockIdx.x * blockDim.x + threadIdx.x) >> 5;
  int lane  = threadIdx.x & 31;
  int m0    = gwave << 4;
  if (m0 >= N) return;
  int mrow = lane & 15, half = lane >> 4;
  int ar = m0 + mrow; if (ar >= N) ar = N - 1;
  const float* arow = Hin + (size_t)ar * HD;
  v8f acc0 = {}, acc1 = {};
#pragma unroll
  for (int k = 0; k < HD; k += 4) {
    int kb = k + half * 2;
    v2f a  = { arow[kb], arow[kb + 1] };
    v2f b0 = { OW[kb * 32 + mrow],      OW[(kb + 1) * 32 + mrow] };
    v2f b1 = { OW[kb * 32 + 16 + mrow], OW[(kb + 1) * 32 + 16 + mrow] };
    acc0 = wmma4(a, b0, acc0);
    acc1 = wmma4(a, b1, acc1);
  }
#pragma unroll
  for (int v = 0; v < 8; v++) {
    int m = m0 + half * 8 + v;
    if (m < N) {
      OUT[(size_t)m * 32 + mrow]      = acc0[v] + OB[mrow];
      OUT[(size_t)m * 32 + 16 + mrow] = acc1[v] + OB[16 + mrow];
    }
  }
}

extern "C" void kernel_launch(void* const* d_in, const int* in_sizes, int n_in,
                              void* d_out, int out_size, void* d_ws, size_t ws_size,
                              hipStream_t stream) {
  const float* x      = (const float*)d_in[0];
  const int*   ei     = (const int*)d_in[1];
  const int*   et     = (const int*)d_in[2];
  const float* ew     = (const float*)d_in[3];
  const float* W1     = (const float*)d_in[4];
  const float* W01    = (const float*)d_in[5];
  const float* alpha1 = (const float*)d_in[6];
  const float* pW1    = (const float*)d_in[7];
  const float* pb1    = (const float*)d_in[8];
  const float* W2     = (const float*)d_in[9];
  const float* W02    = (const float*)d_in[10];
  const float* alpha2 = (const float*)d_in[11];
  const float* pW2    = (const float*)d_in[12];
  const float* pb2    = (const float*)d_in[13];
  const float* outW   = (const float*)d_in[14];
  const float* outb   = (const float*)d_in[15];
  float* out = (float*)d_out;

  int N = in_sizes[0] / HD;
  int E = in_sizes[2];

  float* ws  = (float*)d_ws;
  float* deg = ws;                                  // N  (becomes inv-degree)
  float* Y   = deg + N;                             // 5*N*HD
  float* agg = Y + (size_t)5 * N * HD;              // N*HD
  float* h1  = agg + (size_t)N * HD;                // N*HD
  float* h2  = h1 + (size_t)N * HD;                 // N*HD

  int tiles = (N + 15) / 16;
  dim3 blk(256);
  auto blocks = [](long long threads) {
    return dim3((unsigned)((threads + 255) / 256));
  };

  // degree / inverse degree
  (void)hipMemsetAsync(deg, 0, (size_t)N * sizeof(float), stream);
  k_degree<<<blocks(E), blk, 0, stream>>>(ei, deg, E);
  k_invdeg<<<blocks(N), blk, 0, stream>>>(deg, N);

  // ---- RGCN layer 1 ----
  (void)hipMemsetAsync(agg, 0, (size_t)N * HD * sizeof(float), stream);
  k_xform5<<<blocks((long long)tiles * 5 * 32), blk, 0, stream>>>(x, W1, W01, Y, N);
  k_scatter_rel<<<blocks((long long)E * 16), blk, 0, stream>>>(ei, et, ew, Y, agg, E, N);
  k_rgcn_finalize<<<blocks((long long)N * HD), blk, 0, stream>>>(
      Y + (size_t)4 * N * HD, agg, deg, h1, N);

  // ---- GroupEnhance 1 ----
  (void)hipMemsetAsync(agg, 0, (size_t)N * HD * sizeof(float), stream);
  k_scatter_plain<<<blocks((long long)E * 16), blk, 0, stream>>>(ei, ew, h1, agg, E);
  k_group<<<blocks((long long)tiles * 32), blk, 0, stream>>>(h1, agg, deg, pW1, pb1,
                                                             alpha1, h2, N);

  // ---- RGCN layer 2 ----
  (void)hipMemsetAsync(agg, 0, (size_t)N * HD * sizeof(float), stream);
  k_xform5<<<blocks((long long)tiles * 5 * 32), blk, 0, stream>>>(h2, W2, W02, Y, N);
  k_scatter_rel<<<blocks((long long)E * 16), blk, 0, stream>>>(ei, et, ew, Y, agg, E, N);
  k_rgcn_finalize<<<blocks((long long)N * HD), blk, 0, stream>>>(
      Y + (size_t)4 * N * HD, agg, deg, h1, N);

  // ---- GroupEnhance 2 ----
  (void)hipMemsetAsync(agg, 0, (size_t)N * HD * sizeof(float), stream);
  k_scatter_plain<<<blocks((long long)E * 16), blk, 0, stream>>>(ei, ew, h1, agg, E);
  k_group<<<blocks((long long)tiles * 32), blk, 0, stream>>>(h1, agg, deg, pW2, pb2,
                                                             alpha2, h2, N);

  // ---- output head ----
  k_outhead<<<blocks((long long)tiles * 32), blk, 0, stream>>>(h2, outW, outb, out, N);
}